// SCRFD_ONNX_Wrapper_65678639890614
// MI455X (gfx1250) — compile-verified
//
#include <hip/hip_runtime.h>
#include <stdint.h>

#define L0 12800
#define L1 3200
#define L2 800
#define A_TOT 16800
#define TOPK 512
#define MAXDET 128
#define IMG_F 640.0f
#define IOU_T 0.4f
#define NCHUNK 33   // 33*512 = 16896 >= 16800

typedef int v4i __attribute__((ext_vector_type(4)));
typedef __attribute__((address_space(1))) v4i GV4;   // global 128-bit payload
typedef __attribute__((address_space(3))) v4i LV4;   // LDS 128-bit payload

// key = ordered-float(value) in high 32 bits, ~index in low 32 bits.
// Larger key <=> larger value, ties broken by smaller index.
__device__ __forceinline__ uint64_t make_key(float v, int a) {
  uint32_t u = __float_as_uint(v);
  u = (u & 0x80000000u) ? ~u : (u | 0x80000000u);
  return ((uint64_t)u << 32) | (uint32_t)(0xFFFFFFFFu - (uint32_t)a);
}
__device__ __forceinline__ float key_to_val(uint64_t key) {
  uint32_t u = (uint32_t)(key >> 32);
  uint32_t ob = (u & 0x80000000u) ? (u & 0x7FFFFFFFu) : ~u;
  return __uint_as_float(ob);
}

// intra-wave32 xor-shuffle compare-exchange (pairs j<=16 stay inside a wave)
__device__ __forceinline__ void cmpx_shfl(uint64_t& E, int lane, int j, bool up) {
  uint64_t p = (uint64_t)__shfl_xor((unsigned long long)E, j, 32);
  bool lower = ((lane & j) == 0);
  uint64_t mn = (E < p) ? E : p;
  uint64_t mx = (E < p) ? p : E;
  E = (lower == up) ? mn : mx;
}

// LDS compare-exchange stages j = jstart..32 (k!=0: sort dir by (i&k); k==0: descending)
__device__ __forceinline__ void lds_stages(uint64_t* sh, uint64_t& E0, uint64_t& E1,
                                           int tid, int jstart, int k) {
  sh[tid] = E0; sh[tid + 256] = E1;
  __syncthreads();
  for (int j = jstart; j >= 32; j >>= 1) {
    for (int i = tid; i < TOPK; i += 256) {
      int ixj = i ^ j;
      if (ixj > i) {
        bool up = k ? ((i & k) == 0) : false;
        uint64_t x0 = sh[i], x1 = sh[ixj];
        if ((x0 > x1) == up) { sh[i] = x1; sh[ixj] = x0; }
      }
    }
    __syncthreads();
  }
  E0 = sh[tid]; E1 = sh[tid + 256];
  __syncthreads();
}

// ---------------------------------------------------------------------------
// Kernel 1: per-batch exact top-512 (sorted descending). One block per batch,
// 256 threads. Register-resident running top; async double-buffered chunk
// loads; hybrid shuffle/LDS bitonic network.
// ---------------------------------------------------------------------------
__global__ __launch_bounds__(256) void topk_kernel(
    const float* __restrict__ s8, const float* __restrict__ s16,
    const float* __restrict__ s32,
    uint64_t* __restrict__ keysTop, float* __restrict__ valsTop) {
  __shared__ __align__(16) float raw[2][TOPK];   // double-buffered raw scores
  __shared__ uint64_t sh[TOPK];                  // LDS stage scratch
  const int b = blockIdx.x, tid = threadIdx.x;

  uint64_t S0 = 0ull, S1 = 0ull;   // running top-512: elem[tid], elem[tid+256]

  // issue async b128 load of chunk c's 512 scores into raw[buf]
  auto issue_chunk = [&](int c, int buf) {
    if (tid < 128) {
      int a = c * TOPK + tid * 4;            // 16B-aligned, never straddles a level
      if (a < A_TOT) {
        const float* src;
        if (a < L0)            src = s8  + (size_t)b * L0 + a;
        else if (a < L0 + L1)  src = s16 + (size_t)b * L1 + (a - L0);
        else                   src = s32 + (size_t)b * L2 + (a - L0 - L1);
        __builtin_amdgcn_global_load_async_to_lds_b128(
            (GV4*)src, (LV4*)(&raw[buf][tid * 4]), 0, 0);
      }
    }
  };

  issue_chunk(0, 0);

  for (int c = 0; c < NCHUNK; ++c) {
    if (c + 1 < NCHUNK) {
      issue_chunk(c + 1, (c + 1) & 1);
      __builtin_amdgcn_s_wait_asynccnt(1);   // retire chunk c, keep c+1 in flight
    } else {
      __builtin_amdgcn_s_wait_asynccnt(0);
    }
    __syncthreads();

    // keyify chunk c from LDS into registers
    const float* rc = raw[c & 1];
    int a0 = c * TOPK + tid, a1 = a0 + 256;
    uint64_t E0 = 0ull, E1 = 0ull;
    if (a0 < A_TOT) {
      float sg = 1.0f / (1.0f + expf(-rc[tid]));
      E0 = make_key((sg > 0.5f) ? sg : -1.0f, a0);
    }
    if (a1 < A_TOT) {
      float sg = 1.0f / (1.0f + expf(-rc[tid + 256]));
      E1 = make_key((sg > 0.5f) ? sg : -1.0f, a1);
    }
    __syncthreads();   // raw[c&1] fully consumed before it is reused

    // ---- full ascending bitonic sort of the 512 chunk keys ----
    for (int k = 2; k <= TOPK; k <<= 1) {
      if (k == TOPK) {                       // j=256: intra-thread, ascending
        if (E0 > E1) { uint64_t t = E0; E0 = E1; E1 = t; }
      }
      int jstart = ((k >> 1) > 128) ? 128 : (k >> 1);
      if (jstart >= 32) lds_stages(sh, E0, E1, tid, jstart, k);
      int js = ((k >> 1) > 16) ? 16 : (k >> 1);
      bool up0 = ((tid & k) == 0);
      bool up1 = (((tid + 256) & k) == 0);
      for (int j = js; j >= 1; j >>= 1) {
        cmpx_shfl(E0, tid, j, up0);
        cmpx_shfl(E1, tid, j, up1);
      }
    }

    // ---- S desc + C asc: elementwise max is the top-512 of the union ----
    uint64_t M0 = (S0 > E0) ? S0 : E0;
    uint64_t M1 = (S1 > E1) ? S1 : E1;
    // bitonic merge back to descending
    if (M0 < M1) { uint64_t t = M0; M0 = M1; M1 = t; }   // j=256
    lds_stages(sh, M0, M1, tid, 128, 0);                 // j=128,64,32
    for (int j = 16; j >= 1; j >>= 1) {                  // j=16..1
      cmpx_shfl(M0, tid, j, false);
      cmpx_shfl(M1, tid, j, false);
    }
    S0 = M0; S1 = M1;
  }

  keysTop[(size_t)b * TOPK + tid]       = S0;
  keysTop[(size_t)b * TOPK + tid + 256] = S1;
  valsTop[(size_t)b * TOPK + tid]       = key_to_val(S0);
  valsTop[(size_t)b * TOPK + tid + 256] = key_to_val(S1);
}

// ---------------------------------------------------------------------------
// Kernel 2: decode the 512 selected candidates per batch (anchors on the fly)
// grid = (2, B), block = 256
// ---------------------------------------------------------------------------
__global__ __launch_bounds__(256) void decode_kernel(
    const float* __restrict__ b8,  const float* __restrict__ l8,
    const float* __restrict__ b16, const float* __restrict__ l16,
    const float* __restrict__ b32, const float* __restrict__ l32,
    const uint64_t* __restrict__ keysTop,
    float* __restrict__ cand14, float* __restrict__ boxes4) {
  const int b = blockIdx.y;
  const int k = blockIdx.x * 256 + threadIdx.x;
  uint64_t key = keysTop[(size_t)b * TOPK + k];
  int a = (int)(0xFFFFFFFFu - (uint32_t)(key & 0xFFFFFFFFull));

  const float *bb, *lm; float s; int f, rem;
  if (a < L0)           { bb = b8  + (size_t)b * L0 * 4; lm = l8  + (size_t)b * L0 * 10; s = 8.f;  f = 80; rem = a; }
  else if (a < L0 + L1) { bb = b16 + (size_t)b * L1 * 4; lm = l16 + (size_t)b * L1 * 10; s = 16.f; f = 40; rem = a - L0; }
  else                  { bb = b32 + (size_t)b * L2 * 4; lm = l32 + (size_t)b * L2 * 10; s = 32.f; f = 20; rem = a - L0 - L1; }

  int pos = rem >> 1;                    // anchors repeated 2x per cell
  float cx = ((float)(pos % f) + 0.5f) * s;
  float cy = ((float)(pos / f) + 0.5f) * s;

  const float* bo = bb + (size_t)rem * 4;
  float x1 = (cx - bo[0] * s) / IMG_F;
  float y1 = (cy - bo[1] * s) / IMG_F;
  float x2 = (cx + bo[2] * s) / IMG_F;
  float y2 = (cy + bo[3] * s) / IMG_F;

  float out[14];
  out[0] = x1; out[1] = y1; out[2] = x2; out[3] = y2;
  const float* lp = lm + (size_t)rem * 10;
  #pragma unroll
  for (int j = 0; j < 5; ++j) {
    out[4 + 2 * j] = (lp[2 * j]     * s + cx) / IMG_F;
    out[5 + 2 * j] = (lp[2 * j + 1] * s + cy) / IMG_F;
  }

  size_t base = (size_t)b * TOPK + k;
  #pragma unroll
  for (int j = 0; j < 14; ++j) cand14[base * 14 + j] = out[j];
  ((float4*)boxes4)[base] = make_float4(x1, y1, x2, y2);
}

// ---------------------------------------------------------------------------
// Kernel 3: NMS + compaction + outputs. One block/batch, 512 threads.
// Boxes staged to LDS via CDNA5 async global->LDS b128 path.
// ---------------------------------------------------------------------------
__global__ __launch_bounds__(512) void nms_kernel(
    const float* __restrict__ boxes4, const float* __restrict__ valsTop,
    const float* __restrict__ cand14, float* __restrict__ out, int Bn) {
  __shared__ __align__(16) float bx[TOPK * 4];
  __shared__ float ar[TOPK];
  __shared__ float vv[TOPK];
  __shared__ int   kept[TOPK];
  __shared__ int   rank_[TOPK];
  __shared__ int   totalS;

  const int b = blockIdx.x, tid = threadIdx.x;

  // async stage the 512x4 box tile into LDS
  {
    size_t goff = ((size_t)b * TOPK + tid) * 4;
    __builtin_amdgcn_global_load_async_to_lds_b128(
        (GV4*)(boxes4 + goff), (LV4*)(&bx[tid * 4]), 0, 0);
    __builtin_amdgcn_s_wait_asynccnt(0);
  }
  vv[tid] = valsTop[(size_t)b * TOPK + tid];
  __syncthreads();

  float x1 = bx[tid * 4 + 0], y1 = bx[tid * 4 + 1];
  float x2 = bx[tid * 4 + 2], y2 = bx[tid * 4 + 3];
  ar[tid]   = (x2 - x1) * (y2 - y1);
  kept[tid] = (vv[tid] > 0.5f) ? 1 : 0;
  __syncthreads();

  // sequential greedy suppression, exactly as the reference fori_loop
  for (int i = 0; i < TOPK; ++i) {
    if (kept[i] && tid > i && kept[tid]) {
      float lx = fmaxf(x1, bx[i * 4 + 0]);
      float ly = fmaxf(y1, bx[i * 4 + 1]);
      float rx = fminf(x2, bx[i * 4 + 2]);
      float ry = fminf(y2, bx[i * 4 + 3]);
      float w = fmaxf(rx - lx, 0.0f), h = fmaxf(ry - ly, 0.0f);
      float inter = w * h;
      float iou = inter / (ar[i] + ar[tid] - inter + 1e-9f);
      if (iou > IOU_T) kept[tid] = 0;
    }
    __syncthreads();
  }

  // stable compaction (kept entries are already in descending-score order)
  if (tid == 0) {
    int run = 0;
    for (int i = 0; i < TOPK; ++i) { rank_[i] = run; run += kept[i]; }
    totalS = run;
  }
  __syncthreads();
  int num = totalS; if (num > MAXDET) num = MAXDET;

  // output layout: [num(B)] [boxes(B,128,4)] [scores(B,128)] [lmk(B,128,10)]
  float* o_num = out;
  float* o_box = out + Bn;
  float* o_sc  = out + Bn + (size_t)Bn * MAXDET * 4;
  float* o_lm  = out + Bn + (size_t)Bn * MAXDET * 4 + (size_t)Bn * MAXDET;

  if (kept[tid] && rank_[tid] < MAXDET) {
    int r = rank_[tid];
    const float* c14 = cand14 + ((size_t)b * TOPK + tid) * 14;
    float* db = o_box + ((size_t)b * MAXDET + r) * 4;
    db[0] = c14[0]; db[1] = c14[1]; db[2] = c14[2]; db[3] = c14[3];
    o_sc[(size_t)b * MAXDET + r] = vv[tid];
    float* dl = o_lm + ((size_t)b * MAXDET + r) * 10;
    #pragma unroll
    for (int j = 0; j < 10; ++j) dl[j] = c14[4 + j];
  }
  if (tid < MAXDET && tid >= num) {   // zero-fill unused detection slots
    float* db = o_box + ((size_t)b * MAXDET + tid) * 4;
    db[0] = 0.f; db[1] = 0.f; db[2] = 0.f; db[3] = 0.f;
    o_sc[(size_t)b * MAXDET + tid] = 0.f;
    float* dl = o_lm + ((size_t)b * MAXDET + tid) * 10;
    #pragma unroll
    for (int j = 0; j < 10; ++j) dl[j] = 0.f;
  }
  if (tid == 0) o_num[b] = (float)num;
}

// ---------------------------------------------------------------------------
extern "C" void kernel_launch(void* const* d_in, const int* in_sizes, int n_in,
                              void* d_out, int out_size, void* d_ws, size_t ws_size,
                              hipStream_t stream) {
  const float* s8  = (const float*)d_in[0];
  const float* b8  = (const float*)d_in[1];
  const float* l8  = (const float*)d_in[2];
  const float* s16 = (const float*)d_in[3];
  const float* b16 = (const float*)d_in[4];
  const float* l16 = (const float*)d_in[5];
  const float* s32 = (const float*)d_in[6];
  const float* b32 = (const float*)d_in[7];
  const float* l32 = (const float*)d_in[8];
  const int B = in_sizes[0] / L0;   // 128

  char* w = (char*)d_ws;
  uint64_t* keysTop = (uint64_t*)w;           size_t off = (size_t)B * TOPK * 8;
  float*    valsTop = (float*)(w + off);      off += (size_t)B * TOPK * 4;
  float*    boxes4  = (float*)(w + off);      off += (size_t)B * TOPK * 4 * 4;  // 16B aligned
  float*    cand14  = (float*)(w + off);      // B*TOPK*14 floats

  topk_kernel  <<<B, 256, 0, stream>>>(s8, s16, s32, keysTop, valsTop);
  decode_kernel<<<dim3(2, B), 256, 0, stream>>>(b8, l8, b16, l16, b32, l32,
                                                keysTop, cand14, boxes4);
  nms_kernel   <<<B, 512, 0, stream>>>(boxes4, valsTop, cand14, (float*)d_out, B);
}